// RetinaNet_53575422051039
// MI455X (gfx1250) — compile-verified
//
#include <hip/hip_runtime.h>
#include <stdint.h>

// ---------------- problem constants ----------------
#define A_N   196416
#define C_N   80
#define K_N   1000
#define CAP_N 2048          // candidate buffer per class (>= K + radix-tie slack)
#define NT_N  63            // ceil(1000/16) 16-wide tiles
#define THR   0.05f

typedef __attribute__((ext_vector_type(2))) float v2f;
typedef __attribute__((ext_vector_type(8))) float v8f;

// ---------------- workspace layout (bytes) ----------------
// zero-region first (re-zeroed every launch: accumulated / partially-written buffers)
#define OFF_HIST1   0u                        // C*4096 u32
#define OFF_HIST2   1310720u                  // C*4096 u32
#define OFF_CANDCNT 2621440u                  // C u32
#define OFF_MASK    2621760u                  // C*K*32 u32 suppression bitmap
#define ZERO_U32    3215440u                  // dwords to clear (12,861,760 B)
#define OFF_BOXES   12861760u                 // A float4 decoded boxes
#define OFF_CLST    16004416u                 // C*A f32 transposed scores
#define OFF_SEL1    78857536u                 // C*2 u32 (b1, countAbove)
#define OFF_CUTOFF  78858176u                 // C u32 24-bit cutoff prefix
#define OFF_CAND    78858496u                 // C*CAP uint2 (scorebits, idx)
#define OFF_TOPS    80169216u                 // C*K f32
#define OFF_TOPBOX  80489216u                 // C*K float4
#define OFF_KEEP    81769216u                 // C*K u32

// ---------------- kernels ----------------

__global__ void k_zero(uint32_t* __restrict__ p, int n) {
  int stride = gridDim.x * blockDim.x;
  for (int i = blockIdx.x * blockDim.x + threadIdx.x; i < n; i += stride) p[i] = 0u;
}

// BBoxTransform (std-scaled) + ClipBoxes
__global__ void k_decode(const float4* __restrict__ anc, const float4* __restrict__ reg,
                         float4* __restrict__ boxes) {
  int i = blockIdx.x * blockDim.x + threadIdx.x;
  if (i >= A_N) return;
  float4 an = anc[i], rg = reg[i];
  float wa = an.z - an.x, ha = an.w - an.y;
  float cxa = an.x + 0.5f * wa, cya = an.y + 0.5f * ha;
  float cx = cxa + rg.x * 0.1f * wa;
  float cy = cya + rg.y * 0.1f * ha;
  float w  = expf(rg.z * 0.2f) * wa;
  float h  = expf(rg.w * 0.2f) * ha;
  float4 b;
  b.x = fminf(fmaxf(cx - 0.5f * w, 0.0f), 1024.0f);
  b.y = fminf(fmaxf(cy - 0.5f * h, 0.0f), 1024.0f);
  b.z = fminf(fmaxf(cx + 0.5f * w, 0.0f), 1024.0f);
  b.w = fminf(fmaxf(cy + 0.5f * h, 0.0f), 1024.0f);
  boxes[i] = b;
}

// [A,80] -> [80,A] LDS-tiled transpose so per-class passes stream coalesced
__global__ void k_transpose(const float* __restrict__ cls, float* __restrict__ clsT) {
  __shared__ float tile[64][81];
  int a0 = blockIdx.x * 64;
  for (int t = threadIdx.x; t < 64 * C_N; t += 256) {
    int r = t / C_N, c = t % C_N;
    tile[r][c] = cls[(size_t)(a0 + r) * C_N + c];
  }
  __syncthreads();
  for (int t = threadIdx.x; t < 64 * C_N; t += 256) {
    int c = t / 64, r = t % 64;
    clsT[(size_t)c * A_N + a0 + r] = tile[r][c];
  }
}

// pass 1: 4096-bin histogram of score bits[30:19] (sign always 0 for s>THR)
__global__ void k_hist1(const float* __restrict__ clsT, uint32_t* __restrict__ hist) {
  __shared__ uint32_t h[4096];
  for (int t = threadIdx.x; t < 4096; t += 256) h[t] = 0u;
  __syncthreads();
  int c = blockIdx.x >> 5, chunk = blockIdx.x & 31;
  const float* row = clsT + (size_t)c * A_N;
  for (int a = chunk * 256 + threadIdx.x; a < A_N; a += 32 * 256) {
    __builtin_prefetch(row + a + 32 * 256, 0, 1);   // global_prefetch_b8
    float s = row[a];
    if (s > THR) atomicAdd(&h[(__float_as_uint(s) >> 19) & 0xFFFu], 1u);
  }
  __syncthreads();
  for (int t = threadIdx.x; t < 4096; t += 256) {
    uint32_t v = h[t];
    if (v) atomicAdd(&hist[c * 4096 + t], v);
  }
}

__global__ void k_sel1(const uint32_t* __restrict__ hist, uint32_t* __restrict__ sel) {
  int c = blockIdx.x * blockDim.x + threadIdx.x;
  if (c >= C_N) return;
  uint32_t cum = 0, b1 = 0xFFFFFFFFu, cab = 0;
  for (int b = 4095; b >= 0; --b) {
    uint32_t cnt = hist[c * 4096 + b];
    if (cum + cnt >= K_N) { b1 = (uint32_t)b; cab = cum; break; }
    cum += cnt;
  }
  sel[c * 2] = b1;        // sentinel 0xFFFFFFFF => fewer than K valid, keep all
  sel[c * 2 + 1] = cab;
}

// pass 2: refine inside cutoff bucket on bits[18:7]
__global__ void k_hist2(const float* __restrict__ clsT, const uint32_t* __restrict__ sel,
                        uint32_t* __restrict__ hist) {
  __shared__ uint32_t h[4096];
  for (int t = threadIdx.x; t < 4096; t += 256) h[t] = 0u;
  __syncthreads();
  int c = blockIdx.x >> 5, chunk = blockIdx.x & 31;
  uint32_t b1 = sel[c * 2];
  const float* row = clsT + (size_t)c * A_N;
  for (int a = chunk * 256 + threadIdx.x; a < A_N; a += 32 * 256) {
    __builtin_prefetch(row + a + 32 * 256, 0, 1);
    float s = row[a];
    if (s > THR) {
      uint32_t bits = __float_as_uint(s);
      if (((bits >> 19) & 0xFFFu) == b1) atomicAdd(&h[(bits >> 7) & 0xFFFu], 1u);
    }
  }
  __syncthreads();
  for (int t = threadIdx.x; t < 4096; t += 256) {
    uint32_t v = h[t];
    if (v) atomicAdd(&hist[c * 4096 + t], v);
  }
}

__global__ void k_sel2(const uint32_t* __restrict__ hist2, const uint32_t* __restrict__ sel,
                       uint32_t* __restrict__ cutoff) {
  int c = blockIdx.x * blockDim.x + threadIdx.x;
  if (c >= C_N) return;
  uint32_t b1 = sel[c * 2], cab = sel[c * 2 + 1];
  uint32_t cut = 0u;
  if (b1 != 0xFFFFFFFFu) {
    uint32_t cum = cab, b2 = 0u;
    for (int b = 4095; b >= 0; --b) {
      uint32_t cnt = hist2[c * 4096 + b];
      if (cum + cnt >= K_N) { b2 = (uint32_t)b; break; }
      cum += cnt;
    }
    cut = (b1 << 12) | b2;   // 24-bit prefix threshold
  }
  cutoff[c] = cut;
}

__global__ void k_compact(const float* __restrict__ clsT, const uint32_t* __restrict__ cutoff,
                          uint2* __restrict__ cand, uint32_t* __restrict__ candCnt) {
  int c = blockIdx.x >> 5, chunk = blockIdx.x & 31;
  uint32_t cut = cutoff[c];
  const float* row = clsT + (size_t)c * A_N;
  for (int a = chunk * 256 + threadIdx.x; a < A_N; a += 32 * 256) {
    __builtin_prefetch(row + a + 32 * 256, 0, 1);
    float s = row[a];
    if (s > THR) {
      uint32_t bits = __float_as_uint(s);
      if ((bits >> 7) >= cut) {
        uint32_t pos = atomicAdd(&candCnt[c], 1u);
        if (pos < CAP_N) cand[(size_t)c * CAP_N + pos] = make_uint2(bits, (uint32_t)a);
      }
    }
  }
}

// per-class bitonic sort of <=2048 candidates in LDS, emit top-K + gathered boxes
__global__ void k_sort(const uint2* __restrict__ cand, const uint32_t* __restrict__ candCnt,
                       const float4* __restrict__ boxes,
                       float* __restrict__ topS, float4* __restrict__ topBox) {
  __shared__ unsigned long long keys[CAP_N];
  int c = blockIdx.x;
  uint32_t cnt = candCnt[c]; if (cnt > CAP_N) cnt = CAP_N;
  for (int i = threadIdx.x; i < CAP_N; i += 256) {
    unsigned long long kk = 0ull;
    if (i < (int)cnt) {
      uint2 e = cand[(size_t)c * CAP_N + i];
      kk = ((unsigned long long)e.x << 32) | (unsigned long long)(~e.y); // ties: lower idx first
    }
    keys[i] = kk;
  }
  __syncthreads();
  for (int k = 2; k <= CAP_N; k <<= 1) {
    for (int j = k >> 1; j > 0; j >>= 1) {
      for (int i = threadIdx.x; i < CAP_N; i += 256) {
        int ixj = i ^ j;
        if (ixj > i) {
          unsigned long long a = keys[i], b = keys[ixj];
          bool up = (i & k) == 0;
          bool sw = up ? (a < b) : (a > b);   // overall descending
          if (sw) { keys[i] = b; keys[ixj] = a; }
        }
      }
      __syncthreads();
    }
  }
  for (int q = threadIdx.x; q < K_N; q += 256) {
    unsigned long long key = keys[q];
    uint32_t sbits = (uint32_t)(key >> 32);
    uint32_t idx = ~((uint32_t)key);
    bool valid = sbits != 0u;
    topS[(size_t)c * K_N + q] = valid ? __uint_as_float(sbits) : -1.0f;
    float4 bx = make_float4(0.f, 0.f, 0.f, 0.f);
    if (valid) bx = boxes[idx];
    topBox[(size_t)c * K_N + q] = bx;
  }
}

// Suppression bitmap. One wave per 32-bit mask word (tj pair) per 16-row tile:
//  - area_i + area_j for a 16x16 tile via V_WMMA_F32_16X16X4_F32 (rank-2 outer
//    sum on the matrix pipe; co-executes with the min/max intersection VALU)
//  - row-box broadcasts hoisted out of the tile loop into registers
//  - per-row column masks collected with one __ballot per row-pair (no DS
//    round-trips, no atomics: each word is written exactly once, plain store)
__global__ void k_mask(const float4* __restrict__ topBox, uint32_t* __restrict__ mask) {
  int c  = blockIdx.x / NT_N;
  int ti = blockIdx.x % NT_N;
  int lane = threadIdx.x & 31;
  int wv   = threadIdx.x >> 5;
  int sub  = lane & 15;
  float hsel = (lane < 16) ? 1.0f : 0.0f;
  const float4* tb = topBox + (size_t)c * K_N;

  int gi = 16 * ti + sub;
  int ri = gi < K_N ? gi : (K_N - 1);
  float4 rb = tb[ri];
  float rv = gi < K_N ? 1.0f : 0.0f;
  rb.x *= rv; rb.y *= rv; rb.z *= rv; rb.w *= rv;
  float rowArea = (rb.z - rb.x) * (rb.w - rb.y);
  int mbase = (lane >> 4) * 8;   // D layout: lanes 0-15 -> M=0..7, lanes 16-31 -> M=8..15

  // hoist per-row box broadcasts (fixed for the whole block) into registers
  float hx0[8], hy0[8], hx1[8], hy1[8];
#pragma unroll
  for (int r = 0; r < 8; ++r) {
    int m = mbase + r;
    hx0[r] = __shfl(rb.x, m);
    hy0[r] = __shfl(rb.y, m);
    hx1[r] = __shfl(rb.z, m);
    hy1[r] = __shfl(rb.w, m);
  }

  for (int w = (ti >> 1) + wv; w < 32; w += 8) {       // this wave owns word w
    uint32_t rowWord = 0u;
    int tj0 = (2 * w < ti) ? ti : 2 * w;
    int tj1 = (2 * w + 1 < NT_N) ? (2 * w + 1) : (NT_N - 1);
    for (int tj = tj0; tj <= tj1; ++tj) {
      int gj = 16 * tj + sub;
      int ci = gj < K_N ? gj : (K_N - 1);
      float4 cb = tb[ci];
      float cv = gj < K_N ? 1.0f : 0.0f;
      cb.x *= cv; cb.y *= cv; cb.z *= cv; cb.w *= cv;
      float colArea = (cb.z - cb.x) * (cb.w - cb.y);

      // A(16x4): row m -> [area_m, 1, 0, 0]; B(4x16): col n -> [1, area_n, 0, 0]
      v2f av; av.x = rowArea * hsel; av.y = hsel;
      v2f bv; bv.x = hsel;           bv.y = colArea * hsel;
      v8f acc = {};
      v8f sums = __builtin_amdgcn_wmma_f32_16x16x4_f32(
          false, av, false, bv, (short)0, acc, false, false);

      uint32_t sh = (uint32_t)(tj & 1) << 4;
#pragma unroll
      for (int r = 0; r < 8; ++r) {
        int gim = 16 * ti + mbase + r;
        float ltx = fmaxf(hx0[r], cb.x), lty = fmaxf(hy0[r], cb.y);
        float rbx = fminf(hx1[r], cb.z), rby = fminf(hy1[r], cb.w);
        float iw = fmaxf(rbx - ltx, 0.0f), ih = fmaxf(rby - lty, 0.0f);
        float inter = iw * ih;
        float denom = fmaxf(sums[r] - inter, 1e-8f);   // area_i + area_j - inter
        bool sup = (inter > 0.5f * denom) && (gj > gim) && (gj < K_N);
        // lanes 0-15 vote for row r, lanes 16-31 for row r+8 (same 16 columns)
        uint32_t vote = (uint32_t)__ballot(sup);
        uint32_t half = (sub == r) ? (vote & 0xFFFFu)
                      : (sub == r + 8) ? (vote >> 16) : 0u;
        rowWord |= half << sh;
      }
    }
    if (lane < 16 && gi < K_N)
      mask[((size_t)c * K_N + gi) * 32 + w] = rowWord;  // exclusive owner: plain store
  }
}

// greedy scan: one wave32 per class, lane L owns suppression word L (32*32 = 1024 >= K)
// 4-deep software pipeline keeps mask-row loads in flight across the serial chain.
__global__ void k_nms(const float* __restrict__ topS, const uint32_t* __restrict__ mask,
                      uint32_t* __restrict__ keep) {
  int c = blockIdx.x;
  int lane = threadIdx.x;
  const uint32_t* M = mask + (size_t)c * K_N * 32;
  uint32_t remv = 0u;
  uint32_t p0 = M[lane];
  uint32_t p1 = M[32 + lane];
  uint32_t p2 = M[64 + lane];
  uint32_t p3 = M[96 + lane];
#pragma unroll 4
  for (int i = 0; i < K_N; ++i) {
    uint32_t nxt = (i + 4 < K_N) ? M[(size_t)(i + 4) * 32 + lane] : 0u;
    uint32_t rw = __shfl(remv, i >> 5);
    int removed = (rw >> (i & 31)) & 1;
    int valid = topS[(size_t)c * K_N + i] > THR;
    int kept = valid && !removed;
    if (lane == 0) keep[(size_t)c * K_N + i] = (uint32_t)kept;
    if (kept) remv |= p0;
    p0 = p1; p1 = p2; p2 = p3; p3 = nxt;
  }
}

// outputs in reference tuple order: scores | labels | boxes | keep (all as f32)
__global__ void k_final(const float* __restrict__ topS, const float4* __restrict__ topBox,
                        const uint32_t* __restrict__ keep, float* __restrict__ out) {
  int t = blockIdx.x * blockDim.x + threadIdx.x;
  if (t >= C_N * K_N) return;
  int c = t / K_N;
  uint32_t kp = keep[t];
  out[t] = kp ? topS[t] : 0.0f;
  out[C_N * K_N + t] = (float)c;
  float4 b = topBox[t];
  float* ob = out + 2 * C_N * K_N + (size_t)t * 4;
  ob[0] = kp ? b.x : 0.0f;
  ob[1] = kp ? b.y : 0.0f;
  ob[2] = kp ? b.z : 0.0f;
  ob[3] = kp ? b.w : 0.0f;
  out[6 * C_N * K_N + t] = kp ? 1.0f : 0.0f;
}

// ---------------- launcher ----------------
extern "C" void kernel_launch(void* const* d_in, const int* in_sizes, int n_in,
                              void* d_out, int out_size, void* d_ws, size_t ws_size,
                              hipStream_t stream) {
  (void)in_sizes; (void)n_in; (void)out_size; (void)ws_size;
  const float*  cls = (const float*)d_in[0];     // [1,A,80]
  const float4* reg = (const float4*)d_in[1];    // [1,A,4]
  const float4* anc = (const float4*)d_in[2];    // [1,A,4]

  char* ws = (char*)d_ws;
  uint32_t* hist1   = (uint32_t*)(ws + OFF_HIST1);
  uint32_t* hist2   = (uint32_t*)(ws + OFF_HIST2);
  uint32_t* candCnt = (uint32_t*)(ws + OFF_CANDCNT);
  uint32_t* maskB   = (uint32_t*)(ws + OFF_MASK);
  float4*   boxes   = (float4*)  (ws + OFF_BOXES);
  float*    clsT    = (float*)   (ws + OFF_CLST);
  uint32_t* sel1    = (uint32_t*)(ws + OFF_SEL1);
  uint32_t* cutoff  = (uint32_t*)(ws + OFF_CUTOFF);
  uint2*    cand    = (uint2*)   (ws + OFF_CAND);
  float*    topS    = (float*)   (ws + OFF_TOPS);
  float4*   topBox  = (float4*)  (ws + OFF_TOPBOX);
  uint32_t* keep    = (uint32_t*)(ws + OFF_KEEP);

  k_zero     <<<1024, 256, 0, stream>>>((uint32_t*)ws, (int)ZERO_U32);
  k_decode   <<<(A_N + 255) / 256, 256, 0, stream>>>(anc, reg, boxes);
  k_transpose<<<A_N / 64, 256, 0, stream>>>(cls, clsT);
  k_hist1    <<<C_N * 32, 256, 0, stream>>>(clsT, hist1);
  k_sel1     <<<1, 128, 0, stream>>>(hist1, sel1);
  k_hist2    <<<C_N * 32, 256, 0, stream>>>(clsT, sel1, hist2);
  k_sel2     <<<1, 128, 0, stream>>>(hist2, sel1, cutoff);
  k_compact  <<<C_N * 32, 256, 0, stream>>>(clsT, cutoff, cand, candCnt);
  k_sort     <<<C_N, 256, 0, stream>>>(cand, candCnt, boxes, topS, topBox);
  k_mask     <<<C_N * NT_N, 256, 0, stream>>>(topBox, maskB);
  k_nms      <<<C_N, 32, 0, stream>>>(topS, maskB, keep);
  k_final    <<<(C_N * K_N + 255) / 256, 256, 0, stream>>>(topS, topBox, keep, (float*)d_out);
}